// SimpleGCN_14791867368180
// MI455X (gfx1250) — compile-verified
//
#include <hip/hip_runtime.h>
#include <hip/hip_bf16.h>

typedef __attribute__((ext_vector_type(2))) float v2f;
typedef __attribute__((ext_vector_type(8))) float v8f;

// ---------------------------------------------------------------------------
// Utility fills
// ---------------------------------------------------------------------------
__global__ void fill_kernel(float* __restrict__ p, float v, int n) {
  int i = blockIdx.x * blockDim.x + threadIdx.x;
  if (i < n) p[i] = v;
}

// deg[dst[e]] += 1  (deg pre-initialized to 1.0 for the self-loop)
__global__ void deg_kernel(const int* __restrict__ dst, float* __restrict__ deg, int n_edges) {
  int e = blockIdx.x * blockDim.x + threadIdx.x;
  if (e < n_edges) atomicAdd(&deg[dst[e]], 1.0f);
}

__global__ void rsqrt_kernel(const float* __restrict__ deg, float* __restrict__ dinv, int n) {
  int i = blockIdx.x * blockDim.x + threadIdx.x;
  if (i < n) {
    float d = deg[i];
    dinv[i] = d > 0.0f ? rsqrtf(d) : 0.0f;
  }
}

// ---------------------------------------------------------------------------
// WMMA fp32 GEMM:  out[N x 32] = A[N x KDIM] @ W[KDIM x 32]
// One wave per 16-row tile; two 16x16 accumulators (cols 0-15 / 16-31);
// K stepped by 4 via v_wmma_f32_16x16x4_f32 (exact fp32 semantics).
//
// Fragment layouts per CDNA5 ISA 7.12.2:
//   A (16x4):  lane half = lane>>4, m = lane&15;  vgpr0 = A[m][K=2*half],
//              vgpr1 = A[m][K=2*half+1]
//   B (4x16):  lane n = lane&15; vgpr0 = B[K=2*half][n], vgpr1 = B[K=2*half+1][n]
//   D (16x16): vgpr r -> row (r + 8*half), col n
// ---------------------------------------------------------------------------
template <int KDIM>
__global__ void gemm_wmma_kernel(const float* __restrict__ A,
                                 const float* __restrict__ W,
                                 float* __restrict__ out, int n_rows) {
  const int lane = threadIdx.x & 31;
  const int wid  = (blockIdx.x * blockDim.x + threadIdx.x) >> 5;
  const int row0 = wid << 4;
  if (row0 >= n_rows) return;

  const int half = lane >> 4;
  const int sub  = lane & 15;

  if (row0 + 16 > n_rows) {
    // scalar tail (not hit for N=100000, kept for generality)
    for (int r = row0 + lane; r < n_rows; r += 32) {
      const float* ar = A + (size_t)r * KDIM;
      for (int n = 0; n < 32; ++n) {
        float s = 0.0f;
        for (int k = 0; k < KDIM; ++k) s += ar[k] * W[(size_t)k * 32 + n];
        out[(size_t)r * 32 + n] = s;
      }
    }
    return;
  }

  v8f acc0 = {};
  v8f acc1 = {};
  const float* arow = A + (size_t)(row0 + sub) * KDIM;

#pragma unroll
  for (int k4 = 0; k4 < KDIM; k4 += 4) {
    v2f a;
    a.x = arow[k4 + 2 * half];
    a.y = arow[k4 + 2 * half + 1];

    const float* wr0 = W + (size_t)(k4 + 2 * half) * 32;
    const float* wr1 = wr0 + 32;
    v2f b0, b1;
    b0.x = wr0[sub];
    b0.y = wr1[sub];
    b1.x = wr0[16 + sub];
    b1.y = wr1[16 + sub];

    acc0 = __builtin_amdgcn_wmma_f32_16x16x4_f32(false, a, false, b0,
                                                 (short)0, acc0, false, false);
    acc1 = __builtin_amdgcn_wmma_f32_16x16x4_f32(false, a, false, b1,
                                                 (short)0, acc1, false, false);
  }

  float* obase = out + (size_t)row0 * 32;
#pragma unroll
  for (int r = 0; r < 8; ++r) {
    int row = r + 8 * half;
    obase[(size_t)row * 32 + sub]      = acc0[r];
    obase[(size_t)row * 32 + 16 + sub] = acc1[r];
  }
}

// ---------------------------------------------------------------------------
// Warp-per-edge gather/scale/scatter-add: out[dst] += xw[src] * dinv[s]*dinv[d]
// 32 lanes = 32 features -> one coalesced 128B gather + 32 contiguous
// f32 atomics per edge (all L2-resident: node table is 12.8 MB vs 192 MB L2).
// ---------------------------------------------------------------------------
__global__ void agg_kernel(const float* __restrict__ xw,
                           const int* __restrict__ src,
                           const int* __restrict__ dst,
                           const float* __restrict__ dinv,
                           float* __restrict__ out, int n_edges) {
  const int lane = threadIdx.x & 31;
  const int wid  = (blockIdx.x * blockDim.x + threadIdx.x) >> 5;
  const int nw   = (gridDim.x * blockDim.x) >> 5;
  for (int e = wid; e < n_edges; e += nw) {
    if (e + nw < n_edges) {
      __builtin_prefetch(&src[e + nw], 0, 0);
      __builtin_prefetch(&dst[e + nw], 0, 0);
    }
    int s = src[e];
    int d = dst[e];
    float norm = dinv[s] * dinv[d];
    float v = xw[(size_t)s * 32 + lane] * norm;
    atomicAdd(&out[(size_t)d * 32 + lane], v);
  }
}

// h = relu(agg + xw * dinv^2 + b)   (self-loop term fused here)
__global__ void bias_relu_kernel(const float* __restrict__ agg,
                                 const float* __restrict__ xw,
                                 const float* __restrict__ dinv,
                                 const float* __restrict__ b,
                                 float* __restrict__ out, int n_elems) {
  int i = blockIdx.x * blockDim.x + threadIdx.x;
  if (i >= n_elems) return;
  int node = i >> 5;
  int f    = i & 31;
  float di = dinv[node];
  float v  = agg[i] + xw[i] * di * di + b[f];
  out[i] = v > 0.0f ? v : 0.0f;
}

// out[node] = dot(h[node], Wlin) + blin   (wave32 shuffle reduction)
__global__ void lin_kernel(const float* __restrict__ h,
                           const float* __restrict__ Wlin,
                           const float* __restrict__ blin,
                           float* __restrict__ out, int n_nodes) {
  const int lane = threadIdx.x & 31;
  const int node = (blockIdx.x * blockDim.x + threadIdx.x) >> 5;
  if (node >= n_nodes) return;
  float v = h[(size_t)node * 32 + lane] * Wlin[lane];
#pragma unroll
  for (int off = 16; off > 0; off >>= 1) v += __shfl_down(v, off, 32);
  if (lane == 0) out[node] = v + blin[0];
}

// ---------------------------------------------------------------------------
extern "C" void kernel_launch(void* const* d_in, const int* in_sizes, int n_in,
                              void* d_out, int out_size, void* d_ws, size_t ws_size,
                              hipStream_t stream) {
  const float* x    = (const float*)d_in[0];
  const int*   ei   = (const int*)d_in[1];
  const float* W1   = (const float*)d_in[2];
  const float* b1   = (const float*)d_in[3];
  const float* W2   = (const float*)d_in[4];
  const float* b2   = (const float*)d_in[5];
  const float* Wlin = (const float*)d_in[6];
  const float* blin = (const float*)d_in[7];

  const int N = in_sizes[0] / 64;  // 100000
  const int E = in_sizes[1] / 2;   // 1600000
  const int* src = ei;
  const int* dst = ei + E;

  // Workspace carving (256B aligned): dinv + 3 ping-pong N x 32 buffers.
  char* ws = (char*)d_ws;
  size_t off = 0;
  auto carve = [&](size_t bytes) {
    void* p = ws + off;
    off += (bytes + 255) & ~(size_t)255;
    return p;
  };
  float* dinv = (float*)carve((size_t)N * 4);
  float* bufA = (float*)carve((size_t)N * 32 * 4);  // xw (gemm out)
  float* bufB = (float*)carve((size_t)N * 32 * 4);  // deg, then agg accum
  float* bufC = (float*)carve((size_t)N * 32 * 4);  // h (post-relu)

  const int NF = N * 32;
  const int TB = 256;

  // degrees (self-loop -> init 1.0), then dinv = rsqrt(deg)
  fill_kernel<<<(N + TB - 1) / TB, TB, 0, stream>>>(bufB, 1.0f, N);
  deg_kernel<<<(E + TB - 1) / TB, TB, 0, stream>>>(dst, bufB, E);
  rsqrt_kernel<<<(N + TB - 1) / TB, TB, 0, stream>>>(bufB, dinv, N);

  const int tiles   = (N + 15) / 16;
  const int gblocks = (tiles + 7) / 8;         // 8 waves per 256-thread block
  const int eblocks = (E + 7) / 8;             // warp-per-edge

  // ---- layer 1 ----
  gemm_wmma_kernel<64><<<gblocks, TB, 0, stream>>>(x, W1, bufA, N);
  fill_kernel<<<(NF + TB - 1) / TB, TB, 0, stream>>>(bufB, 0.0f, NF);
  agg_kernel<<<eblocks, TB, 0, stream>>>(bufA, src, dst, dinv, bufB, E);
  bias_relu_kernel<<<(NF + TB - 1) / TB, TB, 0, stream>>>(bufB, bufA, dinv, b1, bufC, NF);

  // ---- layer 2 ----
  gemm_wmma_kernel<32><<<gblocks, TB, 0, stream>>>(bufC, W2, bufA, N);
  fill_kernel<<<(NF + TB - 1) / TB, TB, 0, stream>>>(bufB, 0.0f, NF);
  agg_kernel<<<eblocks, TB, 0, stream>>>(bufA, src, dst, dinv, bufB, E);
  bias_relu_kernel<<<(NF + TB - 1) / TB, TB, 0, stream>>>(bufB, bufA, dinv, b2, bufC, NF);

  // ---- readout ----
  lin_kernel<<<(N * 32 + TB - 1) / TB, TB, 0, stream>>>(bufC, Wlin, blin, (float*)d_out, N);
}